// OriginalKernelizedAttention_4999341932647
// MI455X (gfx1250) — compile-verified
//
#include <hip/hip_runtime.h>

typedef __attribute__((ext_vector_type(2))) float v2f;
typedef __attribute__((ext_vector_type(8))) float v8f;

#define BH     64              // batch * heads
#define SEQ    4096
#define DIM    128             // head_dim == value_dim
#define CHUNKS 8
#define CHUNK  (SEQ / CHUNKS)  // 512 seq positions per workgroup
#define TS     32              // seq-tile staged in LDS per iteration
#define NT     (CHUNK / TS)    // 16 tiles per chunk
#define LDST   132             // padded LDS row stride (floats); 528B, 16B aligned

// Branchless elu(x)+1: exp(min(x,0)) + max(x,0)
//   x>0 : exp(0)+x = x+1 ; x<=0 : exp(x)+0 = exp(x)
__device__ __forceinline__ float phi(float x) {
  return __expf(fminf(x, 0.f)) + fmaxf(x, 0.f);
}

// ---- CDNA5 async global->LDS copy (ASYNCcnt-tracked, no VGPR staging) ----
__device__ __forceinline__ void async_copy_b128(uint32_t lds_byte_off, const void* gaddr) {
  // GV form: LDS byte address taken from a VGPR; global addr = 64-bit VGPR pair
  asm volatile("global_load_async_to_lds_b128 %0, %1, off"
               :: "v"(lds_byte_off), "v"((unsigned long long)(uintptr_t)gaddr)
               : "memory");
}

#if defined(__has_builtin)
#if __has_builtin(__builtin_amdgcn_s_wait_asynccnt)
#define WAIT_ASYNC(n) __builtin_amdgcn_s_wait_asynccnt(n)
#endif
#endif
#ifndef WAIT_ASYNC
#define WAIT_ASYNC(n) asm volatile("s_wait_asynccnt %0" :: "i"(n) : "memory")
#endif

__global__ void zero_ws_kernel(float* __restrict__ p, int n) {
  int i = blockIdx.x * blockDim.x + threadIdx.x;
  if (i < n) p[i] = 0.f;
}

// Phase 1: KV[d,e] += phi(K)^T V over a chunk of S ; Z[d] += sum phi(K)
__global__ __launch_bounds__(256) void phase1_kv(const float* __restrict__ K,
                                                 const float* __restrict__ V,
                                                 float* __restrict__ KV,
                                                 float* __restrict__ Z) {
  const int bh    = blockIdx.y;
  const int chunk = blockIdx.x;
  const float* Kp = K + (size_t)bh * SEQ * DIM;
  const float* Vp = V + (size_t)bh * SEQ * DIM;

  __shared__ float Kt[2][TS * LDST];   // raw K tiles (phi applied at read)
  __shared__ float Vt[2][TS * LDST];   // V tiles

  const int tid   = threadIdx.x;
  const int wave  = tid >> 5;          // 8 waves (wave32)
  const int lane  = tid & 31;
  const int row   = lane & 15;
  const int koff  = (lane >> 4) << 1;  // lane halves carry K={0,1} / K={2,3}
  const int dbase = wave * 16;         // each wave owns one 16-row d-block

  v8f acc[8] = {};
  float zacc = 0.f;                    // Z partial, harvested from A fragments

  const int s_begin = chunk * CHUNK;

  auto issue_tile = [&](int t, int buf) {
    const float4* Kg = (const float4*)(Kp + (size_t)(s_begin + t * TS) * DIM);
    const float4* Vg = (const float4*)(Vp + (size_t)(s_begin + t * TS) * DIM);
    uint32_t kbase = (uint32_t)(uintptr_t)(&Kt[buf][0]);
    uint32_t vbase = (uint32_t)(uintptr_t)(&Vt[buf][0]);
#pragma unroll
    for (int i = 0; i < 4; ++i) {
      int idx = tid + i * 256;         // 1024 float4 per 32x128 tile
      int r   = idx >> 5;              // 32 float4 per row
      int c4  = idx & 31;
      uint32_t dst = (uint32_t)((r * LDST + c4 * 4) * 4);
      async_copy_b128(kbase + dst, Kg + idx);
      async_copy_b128(vbase + dst, Vg + idx);
    }
  };

  issue_tile(0, 0);                    // prologue: tile 0 in flight (8 ops)

  for (int t = 0; t < NT; ++t) {
    if (t + 1 < NT) {
      issue_tile(t + 1, (t + 1) & 1);  // prefetch next tile (8 more ops)
      WAIT_ASYNC(8);                   // in-order completion: tile t done
    } else {
      WAIT_ASYNC(0);
    }
    __syncthreads();                   // all waves' stages for tile t visible

    const float* Ktc = Kt[t & 1];
    const float* Vtc = Vt[t & 1];

    // WMMA over the 32 staged seq positions, K=4 per instruction.
    // A-fragments cover each (k,d) element of the tile exactly once per WG,
    // so Z is accumulated here for free (no second pass over the K tile).
#pragma unroll
    for (int kk = 0; kk < TS; kk += 4) {
      v2f a;  // A[M=d][k] = phi(K)[s+kk+k][dbase+M]
      a.x = phi(Ktc[(kk + koff) * LDST + dbase + row]);
      a.y = phi(Ktc[(kk + koff + 1) * LDST + dbase + row]);
      zacc += a.x + a.y;
#pragma unroll
      for (int n = 0; n < 8; ++n) {
        v2f b;  // B[k][N=e] = V[s+kk+k][16n+N]
        b.x = Vtc[(kk + koff) * LDST + n * 16 + row];
        b.y = Vtc[(kk + koff + 1) * LDST + n * 16 + row];
        acc[n] = __builtin_amdgcn_wmma_f32_16x16x4_f32(
            false, a, false, b, (short)0, acc[n], false, false);
      }
    }
    __syncthreads();                   // reads done before buffer is re-filled
  }

  // Merge chunk partials (workspace lives in the 192MB L2)
  float* KVp = KV + (size_t)bh * DIM * DIM;
#pragma unroll
  for (int n = 0; n < 8; ++n) {
#pragma unroll
    for (int r = 0; r < 8; ++r) {
      int M = ((lane >> 4) << 3) + r;  // C layout: lanes>=16 hold M=8..15
      unsafeAtomicAdd(&KVp[(dbase + M) * DIM + n * 16 + row], acc[n][r]);
    }
  }
  // Combine the two koff halves: d = dbase + row owned by this wave only
  zacc += __shfl_xor(zacc, 16, 32);
  if (lane < 16) unsafeAtomicAdd(&Z[bh * DIM + dbase + lane], zacc);
}

// Phase 2: out[q,e] = (phi(Q) @ KV)[q,e] / (phi(Q)[q,:] . Z + eps)
__global__ __launch_bounds__(256) void phase2_out(const float* __restrict__ Q,
                                                  const float* __restrict__ KV,
                                                  const float* __restrict__ Z,
                                                  float* __restrict__ out) {
  const int bh     = blockIdx.y;
  const float* Qp  = Q  + (size_t)bh * SEQ * DIM;
  const float* KVp = KV + (size_t)bh * DIM * DIM;
  const float* Zp  = Z  + bh * DIM;
  float* Op        = out + (size_t)bh * SEQ * DIM;

  __shared__ float KVt[DIM * LDST];    // 67.6 KB of the 320 KB WGP LDS

  const int tid  = threadIdx.x;
  const int wave = tid >> 5;
  const int lane = tid & 31;
  const int row  = lane & 15;
  const int koff = (lane >> 4) << 1;
  const int qb   = blockIdx.x * 128 + wave * 16;  // 16 q-rows per wave

  // Kick off async staging of KV (64 KB, L2-hot): 4096 float4, 16 per thread.
  {
    uint32_t base = (uint32_t)(uintptr_t)(&KVt[0]);
#pragma unroll
    for (int i = 0; i < 16; ++i) {
      int idx = tid + i * 256;
      int r   = idx >> 5;
      int c4  = idx & 31;
      uint32_t dst = (uint32_t)((r * LDST + c4 * 4) * 4);
      async_copy_b128(base + dst, ((const float4*)KVp) + idx);
    }
  }

  // Overlap: preload all A fragments (K-dim 128 -> 32 steps of K=4, 64 VGPRs)
  v2f a[32];
  const float* qrow = Qp + (size_t)(qb + row) * DIM;
#pragma unroll
  for (int s = 0; s < 32; ++s) {
    int k0 = 4 * s + koff;
    a[s].x = phi(qrow[k0]);
    a[s].y = phi(qrow[k0 + 1]);
  }

  // Denominator: each lane covers K = {koff, koff+1} mod 4; xor-16 completes it
  float den = 0.f;
#pragma unroll
  for (int s = 0; s < 32; ++s) {
    int k0 = 4 * s + koff;
    den += a[s].x * Zp[k0] + a[s].y * Zp[k0 + 1];
  }
  den += __shfl_xor(den, 16, 32);      // lanes L and L+16 both hold den[qb+(L&15)]

  WAIT_ASYNC(0);
  __syncthreads();

#pragma unroll
  for (int n = 0; n < 8; ++n) {
    v8f acc = {};
#pragma unroll
    for (int s = 0; s < 32; ++s) {
      int k0 = 4 * s + koff;
      v2f b;  // B[k=d][N=e] = KV[k0+..][16n+N]
      b.x = KVt[k0 * LDST + n * 16 + row];
      b.y = KVt[(k0 + 1) * LDST + n * 16 + row];
      acc = __builtin_amdgcn_wmma_f32_16x16x4_f32(
          false, a[s], false, b, (short)0, acc, false, false);
    }
#pragma unroll
    for (int r = 0; r < 8; ++r) {
      int M = ((lane >> 4) << 3) + r;
      float dq = __shfl(den, M, 32);   // den for q = qb + M
      Op[(size_t)(qb + M) * DIM + n * 16 + row] = acc[r] / (dq + 1e-6f);
    }
  }
}

extern "C" void kernel_launch(void* const* d_in, const int* in_sizes, int n_in,
                              void* d_out, int out_size, void* d_ws, size_t ws_size,
                              hipStream_t stream) {
  const float* Q = (const float*)d_in[0];
  const float* K = (const float*)d_in[1];
  const float* V = (const float*)d_in[2];
  float* outp = (float*)d_out;

  // Workspace: KV [BH][128][128] (4 MB) then Z [BH][128] (32 KB)
  float* KV = (float*)d_ws;
  float* Z  = KV + (size_t)BH * DIM * DIM;
  int nz = BH * DIM * DIM + BH * DIM;

  zero_ws_kernel<<<(nz + 255) / 256, 256, 0, stream>>>(KV, nz);
  phase1_kv<<<dim3(CHUNKS, BH), 256, 0, stream>>>(K, V, KV, Z);
  phase2_out<<<dim3(SEQ / 128, BH), 256, 0, stream>>>(Q, KV, Z, outp);
}